// LDConv_22634477649940
// MI455X (gfx1250) — compile-verified
//
#include <hip/hip_runtime.h>
#include <math.h>

typedef float v2f __attribute__((ext_vector_type(2)));
typedef float v8f __attribute__((ext_vector_type(8)));

#define B_  16
#define C_  64
#define H_  128
#define W_  128
#define NP  5
#define K1  320   // N*C
#define C2  128   // 2C
#define CO  64
#define TM  64    // pixels per tile (row segment)

// ---- LDS layout (float offsets). Region A is phase-overlaid. ----
#define OFF_XPATCH 0          // [64 ch][3 rows][66 cols] = 12672   (phase A/B)
#define OFF_OW     12672      // 10*64*9 = 5760                     (phase A/B)
#define OFF_OB     18432      // 10                                 (phase A/B)
#define OFF_W1C    0          // [128][68] = 8704                   (phase C, overlays xpatch)
#define OFF_W2C    8704       // [64][132] = 8448                   (phase D, overlays)
#define OFF_COORDS 18442      // [5][64][2] = 640
#define OFF_FEATS  19082      // [64][68] = 4352 (feats chunk, later out-transpose)
#define OFF_H      23434      // [64][132] = 8448
#define LDS_FLOATS 31882      // 127528 bytes

extern "C" __global__ __launch_bounds__(256)
void ldconv_fused(const float* __restrict__ x,  const float* __restrict__ ow,
                  const float* __restrict__ ob, const float* __restrict__ w1,
                  const float* __restrict__ b1, const float* __restrict__ w2,
                  const float* __restrict__ scale_p, float* __restrict__ out)
{
  extern __shared__ float lds[];
  const int t   = threadIdx.x;
  const int bid = blockIdx.x;
  const int w0  = (bid & 1) * TM;        // W/TM = 2 tiles per row
  const int h   = (bid >> 1) & (H_ - 1);
  const int b   = bid >> 8;              // 256 tiles per batch image
  const float scale = scale_p[0];

  // ---------- Phase A: stage x halo patch (3 x 66 x 64ch) + offset conv weights ----------
  for (int f = t; f < 64 * 3 * 66; f += 256) {
    const int c  = f / 198;
    const int r  = (f % 198) / 66;
    const int cc = f % 66;
    const int hh = h - 1 + r;
    const int ww = w0 - 1 + cc;
    float v = 0.f;
    if (hh >= 0 && hh < H_ && ww >= 0 && ww < W_)
      v = x[((b * C_ + c) * H_ + hh) * W_ + ww];
    lds[OFF_XPATCH + f] = v;   // layout c*198 + r*66 + cc == f
  }
  for (int f = t; f < 5760; f += 256) lds[OFF_OW + f] = ow[f];
  if (t < 10) lds[OFF_OB + t] = ob[t];
  __syncthreads();

  // ---------- Phase B: 3x3 offset conv -> tanh*scale -> pixel sample coords ----------
  for (int p = t; p < 640; p += 256) {          // 10 offset ch x 64 px
    const int oc = p >> 6;
    const int m  = p & 63;
    float acc = lds[OFF_OB + oc];
    const float* xp = &lds[OFF_XPATCH + m];
    const float* wp = &lds[OFF_OW + oc * 576];
    for (int c = 0; c < 64; ++c) {
      const float* xr = xp + c * 198;
      const float* wr = wp + c * 9;
      #pragma unroll
      for (int r = 0; r < 3; ++r) {
        acc = fmaf(xr[r * 66 + 0], wr[r * 3 + 0], acc);
        acc = fmaf(xr[r * 66 + 1], wr[r * 3 + 1], acc);
        acc = fmaf(xr[r * 66 + 2], wr[r * 3 + 2], acc);
      }
    }
    const int   n = oc >> 1;
    const float d = tanhf(acc) * scale;
    if ((oc & 1) == 0) {   // x component (grid[...,0])
      const float gn = -1.f + 2.f * (float)(w0 + m) / (float)(W_ - 1) + d;
      lds[OFF_COORDS + (n * 64 + m) * 2 + 0] = (gn + 1.f) * 0.5f * (float)(W_ - 1);
    } else {               // y component
      const float gn = -1.f + 2.f * (float)h / (float)(H_ - 1) + d;
      lds[OFF_COORDS + (n * 64 + m) * 2 + 1] = (gn + 1.f) * 0.5f * (float)(H_ - 1);
    }
  }

  // ---------- Phase C: for each of 5 offsets: sample feats chunk + GEMM1 accumulate ----------
  const int wv   = t >> 5;            // wave id 0..7 (wave32)
  const int lane = t & 31;
  const int lm   = lane & 15;         // M/N within 16x16 tile
  const int kh   = (lane >> 4) * 2;   // K sub-offset: lanes 16-31 hold K=2,3
  const int mt   = wv & 3;            // 4 M-tiles (64 px)
  const int ntb  = (wv >> 2) * 4;     // each wave: 4 consecutive N-tiles of 8

  v8f acc1[4];
  #pragma unroll
  for (int i = 0; i < 4; ++i)
    acc1[i] = (v8f){0.f, 0.f, 0.f, 0.f, 0.f, 0.f, 0.f, 0.f};

  const int m_s = t & 63;   // sampling pixel for this thread
  const int cb  = t >> 6;   // sampling channel base (0..3)

  for (int n = 0; n < NP; ++n) {
    __syncthreads();  // protect LDS restage vs previous iteration's WMMA reads
    // bilinear sample: feats[m][c] for channels of offset group n (zeros padding)
    {
      const float gx  = lds[OFF_COORDS + (n * 64 + m_s) * 2 + 0];
      const float gy  = lds[OFF_COORDS + (n * 64 + m_s) * 2 + 1];
      const float x0f = floorf(gx), y0f = floorf(gy);
      const int   ix0 = (int)x0f, iy0 = (int)y0f;
      const int   ix1 = ix0 + 1,  iy1 = iy0 + 1;
      const float wx1 = gx - x0f, wx0 = 1.f - wx1;
      const float wy1 = gy - y0f, wy0 = 1.f - wy1;
      const bool vx0 = (ix0 >= 0) & (ix0 < W_);
      const bool vx1 = (ix1 >= 0) & (ix1 < W_);
      const bool vy0 = (iy0 >= 0) & (iy0 < H_);
      const bool vy1 = (iy1 >= 0) & (iy1 < H_);
      const int cx0 = min(max(ix0, 0), W_ - 1), cx1 = min(max(ix1, 0), W_ - 1);
      const int cy0 = min(max(iy0, 0), H_ - 1), cy1 = min(max(iy1, 0), H_ - 1);
      const float w00 = (vy0 && vx0) ? wy0 * wx0 : 0.f;
      const float w01 = (vy0 && vx1) ? wy0 * wx1 : 0.f;
      const float w10 = (vy1 && vx0) ? wy1 * wx0 : 0.f;
      const float w11 = (vy1 && vx1) ? wy1 * wx1 : 0.f;
      const int i00 = cy0 * W_ + cx0, i01 = cy0 * W_ + cx1;
      const int i10 = cy1 * W_ + cx0, i11 = cy1 * W_ + cx1;
      for (int c = cb; c < 64; c += 4) {
        const float* bp = x + (size_t)((b * C_ + c) * H_) * W_;
        const float v = w00 * bp[i00] + w01 * bp[i01] + w10 * bp[i10] + w11 * bp[i11];
        lds[OFF_FEATS + m_s * 68 + c] = v;
      }
    }
    // stage w1 K-slice: w1c[o][k] = w1[o, n*64 + k]  (B matrix: B[k][o] = w1[o][k])
    for (int f = t; f < 128 * 64; f += 256) {
      const int o = f >> 6, k = f & 63;
      lds[OFF_W1C + o * 68 + k] = w1[o * K1 + n * 64 + k];
    }
    __syncthreads();
    // 16 K-steps of 4: D = A(16x4) x B(4x16) + C
    const float* fA = &lds[OFF_FEATS + (mt * 16 + lm) * 68 + kh];
    #pragma unroll 4
    for (int kk = 0; kk < 16; ++kk) {
      const v2f a = *(const v2f*)(fA + kk * 4);
      #pragma unroll
      for (int i = 0; i < 4; ++i) {
        const v2f bb = *(const v2f*)(&lds[OFF_W1C + ((ntb + i) * 16 + lm) * 68 + kk * 4 + kh]);
        acc1[i] = __builtin_amdgcn_wmma_f32_16x16x4_f32(
            false, a, false, bb, (short)0, acc1[i], false, false);
      }
    }
  }
  __syncthreads();

  // ---------- GEMM1 epilogue: h = relu(acc + b1) -> LDS ; stage w2 ----------
  const int mhi = (lane >> 4) * 8;   // C/D layout: lanes 16-31 hold M = r+8
  #pragma unroll
  for (int i = 0; i < 4; ++i) {
    const int   n0   = (ntb + i) * 16 + lm;
    const float bias = b1[n0];
    #pragma unroll
    for (int r = 0; r < 8; ++r) {
      const int M = mt * 16 + mhi + r;
      lds[OFF_H + M * 132 + n0] = fmaxf(acc1[i][r] + bias, 0.f);
    }
  }
  for (int f = t; f < 64 * 128; f += 256) {
    const int o = f >> 7, k = f & 127;
    lds[OFF_W2C + o * 132 + k] = w2[f];   // w2 is [64 out][128 in]
  }
  __syncthreads();

  // ---------- GEMM2: out(64px x 64ch) = h(64x128) x w2^T ----------
  const int nt2b = (wv >> 2) * 2;
  v8f acc2[2];
  #pragma unroll
  for (int i = 0; i < 2; ++i)
    acc2[i] = (v8f){0.f, 0.f, 0.f, 0.f, 0.f, 0.f, 0.f, 0.f};

  const float* hA = &lds[OFF_H + (mt * 16 + lm) * 132 + kh];
  #pragma unroll 4
  for (int kk = 0; kk < 32; ++kk) {
    const v2f a = *(const v2f*)(hA + kk * 4);
    #pragma unroll
    for (int i = 0; i < 2; ++i) {
      const v2f bb = *(const v2f*)(&lds[OFF_W2C + ((nt2b + i) * 16 + lm) * 132 + kk * 4 + kh]);
      acc2[i] = __builtin_amdgcn_wmma_f32_16x16x4_f32(
          false, a, false, bb, (short)0, acc2[i], false, false);
    }
  }

  // transpose result through LDS (reuse feats region) for coalesced stores
  #pragma unroll
  for (int i = 0; i < 2; ++i) {
    const int n0 = (nt2b + i) * 16 + lm;   // output channel
    #pragma unroll
    for (int r = 0; r < 8; ++r) {
      const int M = mt * 16 + mhi + r;     // pixel within tile
      lds[OFF_FEATS + n0 * 68 + M] = acc2[i][r];
    }
  }
  __syncthreads();

  for (int f = t; f < 64 * 64; f += 256) {
    const int co = f >> 6, m = f & 63;
    out[((b * CO + co) * H_ + h) * W_ + w0 + m] = lds[OFF_FEATS + co * 68 + m];
  }
}

extern "C" void kernel_launch(void* const* d_in, const int* in_sizes, int n_in,
                              void* d_out, int out_size, void* d_ws, size_t ws_size,
                              hipStream_t stream) {
  (void)in_sizes; (void)n_in; (void)out_size; (void)d_ws; (void)ws_size;
  const float* x  = (const float*)d_in[0];
  const float* ow = (const float*)d_in[1];
  const float* ob = (const float*)d_in[2];
  const float* w1 = (const float*)d_in[3];
  const float* b1 = (const float*)d_in[4];
  const float* w2 = (const float*)d_in[5];
  const float* sc = (const float*)d_in[6];
  float* out = (float*)d_out;

  const size_t shmem = (size_t)LDS_FLOATS * sizeof(float);  // ~127.5 KB of the 320 KB WGP LDS
  (void)hipFuncSetAttribute((const void*)ldconv_fused,
                            hipFuncAttributeMaxDynamicSharedMemorySize, (int)shmem);

  const int nblocks = B_ * H_ * (W_ / TM);   // 16 * 128 * 2 = 4096
  ldconv_fused<<<dim3(nblocks), dim3(256), shmem, stream>>>(x, ow, ob, w1, b1, w2, sc, out);
}